// ODEfunc_7267084665331
// MI455X (gfx1250) — compile-verified
//
#include <hip/hip_runtime.h>

typedef float v2f __attribute__((ext_vector_type(2)));
typedef float v8f __attribute__((ext_vector_type(8)));

#define B_SZ 4096
#define D_SZ 128
#define H_SZ 512
#define MT   32     // rows per workgroup

// Branchless tanh: prefer CDNA5 V_TANH; else 1 - 2/(1+exp2(2*log2e*x))
// (v_exp_f32 saturates: x->+inf => 1, x->-inf => -1; no EXEC divergence)
__device__ __forceinline__ float fast_tanh(float x) {
#if __has_builtin(__builtin_amdgcn_tanhf)
    return __builtin_amdgcn_tanhf(x);
#else
    float e = __builtin_amdgcn_exp2f(x * 2.8853900817779268f);
    return 1.f - 2.f * __builtin_amdgcn_rcpf(1.f + e);
#endif
}

// K1: s[k] = dot(W1[:,k], W2[k,:])  (closed-form trace helper)
__global__ void prep_s(const float* __restrict__ W1, const float* __restrict__ W2,
                       float* __restrict__ s) {
    int k = blockIdx.x * blockDim.x + threadIdx.x;
    if (k >= H_SZ) return;
    float acc = 0.f;
#pragma unroll 8
    for (int i = 0; i < D_SZ; ++i)
        acc += W1[i * H_SZ + k] * W2[k * D_SZ + i];
    s[k] = acc;
}

// K2: fused  Z = y@W1 ; h = tanh(Z + t*v1 + b1) ; dy = h@W2 + b2 ;
//     div partials = sum_k (1-h^2)*s[k]
__global__ __launch_bounds__(256)
void ode_fused(const float* __restrict__ tptr,
               const float* __restrict__ y,
               const float* __restrict__ W1,
               const float* __restrict__ b1,
               const float* __restrict__ v1,
               const float* __restrict__ W2,
               const float* __restrict__ b2,
               const float* __restrict__ s,
               float* __restrict__ dy,
               float* __restrict__ part) {
    __shared__ float hLDS[MT * H_SZ];           // 64 KB: staged activations
    const float t   = tptr[0];
    const int lane  = threadIdx.x & 31;
    const int wave  = threadIdx.x >> 5;         // 0..7
    const int lhalf = lane >> 4;                // 0 or 1
    const int l16   = lane & 15;
    const int rowBase = blockIdx.x * MT;

    // ---------------- GEMM1: (MT x 128) @ (128 x 512) ----------------
    // 64 output tiles (2 mt x 32 nt); wave owns fixed mt, 8 nt tiles.
    const int mt   = wave >> 2;
    const int rowA = rowBase + mt * 16 + l16;   // A-matrix row for this lane
    float divacc[8];
#pragma unroll
    for (int j = 0; j < 8; ++j) divacc[j] = 0.f;

    for (int i = 0; i < 8; ++i) {
        const int nt  = (wave & 3) * 8 + i;
        const int col = nt * 16 + l16;
        v8f acc = {};
#pragma unroll 8
        for (int k0 = 0; k0 < D_SZ; k0 += 4) {
            const int kk = k0 + 2 * lhalf;
            v2f a = *(const v2f*)(y + rowA * D_SZ + kk);
            v2f b; b[0] = W1[kk * H_SZ + col]; b[1] = W1[(kk + 1) * H_SZ + col];
            acc = __builtin_amdgcn_wmma_f32_16x16x4_f32(
                false, a, false, b, (short)0, acc, false, false);
        }
        const float add = t * v1[col] + b1[col];
        const float sc  = s[col];
#pragma unroll
        for (int j = 0; j < 8; ++j) {
            const float h = fast_tanh(acc[j] + add);
            const int rloc = mt * 16 + j + 8 * lhalf;
            hLDS[rloc * H_SZ + col] = h;
            divacc[j] += (1.f - h * h) * sc;
        }
    }
    // reduce divergence partials across the 16 lanes sharing each row
#pragma unroll
    for (int j = 0; j < 8; ++j) {
        float v = divacc[j];
        for (int m = 1; m < 16; m <<= 1) v += __shfl_xor(v, m, 32);
        divacc[j] = v;
    }
    if (l16 == 0) {
#pragma unroll
        for (int j = 0; j < 8; ++j) {
            const int rg = rowBase + mt * 16 + j + 8 * lhalf;
            part[rg * 4 + (wave & 3)] = divacc[j];   // deterministic, no atomics
        }
    }
    __syncthreads();

    // ---------------- GEMM2: (MT x 512) @ (512 x 128) ----------------
    // 16 output tiles (2 mt x 8 nt); 2 tiles per wave.
    for (int i = 0; i < 2; ++i) {
        const int tile = wave * 2 + i;
        const int mt2 = tile >> 3, nt2 = tile & 7;
        const int col = nt2 * 16 + l16;
        const int rA  = mt2 * 16 + l16;
        v8f acc = {};
#pragma unroll 8
        for (int k0 = 0; k0 < H_SZ; k0 += 4) {
            const int kk = k0 + 2 * lhalf;
            v2f a = *(const v2f*)(&hLDS[rA * H_SZ + kk]);
            v2f b; b[0] = W2[kk * D_SZ + col]; b[1] = W2[(kk + 1) * D_SZ + col];
            acc = __builtin_amdgcn_wmma_f32_16x16x4_f32(
                false, a, false, b, (short)0, acc, false, false);
        }
        const float bb = b2[col];
#pragma unroll
        for (int j = 0; j < 8; ++j) {
            const int rg = rowBase + mt2 * 16 + j + 8 * lhalf;
            dy[rg * D_SZ + col] = acc[j] + bb;
        }
    }
}

// K3: out_div[r] = -(sum of 4 per-wave partials)
__global__ void finish_div(const float* __restrict__ part, float* __restrict__ out_div) {
    int r = blockIdx.x * blockDim.x + threadIdx.x;
    if (r >= B_SZ) return;
    out_div[r] = -(part[4 * r] + part[4 * r + 1] + part[4 * r + 2] + part[4 * r + 3]);
}

extern "C" void kernel_launch(void* const* d_in, const int* in_sizes, int n_in,
                              void* d_out, int out_size, void* d_ws, size_t ws_size,
                              hipStream_t stream) {
    // setup_inputs order: t, y, logp, W1, b1, v1, W2, b2
    const float* t  = (const float*)d_in[0];
    const float* y  = (const float*)d_in[1];
    const float* W1 = (const float*)d_in[3];
    const float* b1 = (const float*)d_in[4];
    const float* v1 = (const float*)d_in[5];
    const float* W2 = (const float*)d_in[6];
    const float* b2 = (const float*)d_in[7];

    float* out_dy  = (float*)d_out;                 // (B, D)
    float* out_div = out_dy + (size_t)B_SZ * D_SZ;  // (B, 1)

    float* s    = (float*)d_ws;      // 512 floats
    float* part = s + H_SZ;          // 4096*4 floats

    prep_s   <<<(H_SZ + 255) / 256, 256, 0, stream>>>(W1, W2, s);
    ode_fused<<<B_SZ / MT, 256, 0, stream>>>(t, y, W1, b1, v1, W2, b2, s, out_dy, part);
    finish_div<<<B_SZ / 256, 256, 0, stream>>>(part, out_div);
}